// LSTMOnline_22763326668968
// MI455X (gfx1250) — compile-verified
//
#include <hip/hip_runtime.h>
#include <math.h>

// LSTM online step, batch=1, E=1024, H=4096, fp32.
//
// Roofline: ~545 MB of weights touched once, ~285 MFLOP -> 0.5 FLOP/byte.
// Pure HBM-bandwidth bound (~23 us floor at 23.3 TB/s). Design:
//  - stream fp32 weights once: per-lane global_load_b64 A tiles, clause'd,
//    unroll 16 -> ~4 KB outstanding per wave
//  - maximize waves in flight (BDP at 23.3 TB/s is ~10+ MB): 8 waves per
//    workgroup split K, W_ih/W_hh tiles are independent workgroups; partial
//    sums reduced deterministically through LDS (no fp atomics)
//  - shared vectors (x, xi, h0, h1) staged in LDS, read via ds_load_2addr_b64
//  - 16-row K-chunk GEMV via V_WMMA_F32_16X16X4_F32 (exact fp32, cross-lane
//    K-reduction for free)

#define E_DIM 1024
#define H_DIM 4096
#define NWAVES 8          // waves per workgroup (256 threads, wave32)

typedef float v2f __attribute__((ext_vector_type(2)));
typedef float v8f __attribute__((ext_vector_type(8)));

// One wave: acc[m] = W[row0+m, k0:k0+KCHUNK] . v[k0:k0+KCHUNK], m = 0..15.
// A layout (32-bit 16x4, ISA 7.12.2): lanes 0-15 hold K=0,1 in VGPR0,1;
// lanes 16-31 hold K=2,3. B = v chunk broadcast across all 16 N columns,
// same K striping (VGPR r, half h -> K = 2h + r).
template <int KCHUNK>
__device__ __forceinline__ v8f wmma_gemv_partial(const float* __restrict__ W,
                                                 int ldw, int row0, int k0,
                                                 const float* __restrict__ v) {
  const int lane = threadIdx.x & 31;
  const int half = lane >> 4;  // 0 or 1
  const int m    = lane & 15;
  const float* wp = W + (size_t)(row0 + m) * (size_t)ldw + k0 + 2 * half;
  const float* vp = v + k0 + 2 * half;
  v8f acc = {};
#pragma unroll 16
  for (int k = 0; k < KCHUNK; k += 4) {
    v2f a = *(const v2f*)(wp + k);  // global_load_b64: 16 rows x 4 K / wave
    v2f b = *(const v2f*)(vp + k);  // ds_load_b64 from LDS
    acc = __builtin_amdgcn_wmma_f32_16x16x4_f32(
        /*neg_a=*/false, a, /*neg_b=*/false, b,
        /*c_mod=*/(short)0, acc, /*reuse_a=*/false, /*reuse_b=*/false);
  }
  return acc;
}

// D layout: VGPR i -> M=i (lanes 0-15), M=8+i (lanes 16-31); all N columns
// equal (B broadcast). Each wave parks its 16 partial row-sums in LDS, then
// threads 0..15 reduce across waves and store coalesced (+ optional bias).
template <bool HAS_BIAS>
__device__ __forceinline__ void reduce_store(float* __restrict__ dst, int row0,
                                             v8f acc,
                                             const float* __restrict__ bias,
                                             float* __restrict__ sred) {
  const int lane = threadIdx.x & 31;
  const int wave = threadIdx.x >> 5;
  if ((lane & 15) == 0) {
    const int mbase = (lane >> 4) * 8;
#pragma unroll
    for (int i = 0; i < 8; ++i) sred[wave * 16 + mbase + i] = acc[i];
  }
  __syncthreads();
  const int t = threadIdx.x;
  if (t < 16) {
    float s = 0.0f;
#pragma unroll
    for (int w = 0; w < NWAVES; ++w) s += sred[w * 16 + t];
    if (HAS_BIAS) s += bias[row0 + t];
    dst[row0 + t] = s;
  }
}

// ---- Stage 1: xi = W_in @ x + b_in  (H rows, K=E) ----
__global__ void __launch_bounds__(32 * NWAVES)
k_in_proj(const float* __restrict__ x, const float* __restrict__ W_in,
          const float* __restrict__ b_in, float* __restrict__ xi) {
  __shared__ float sx[E_DIM];
  __shared__ float sred[NWAVES * 16];
  for (int i = threadIdx.x; i < E_DIM; i += 32 * NWAVES) sx[i] = x[i];
  __syncthreads();
  constexpr int KC = E_DIM / NWAVES;  // 128
  const int row0 = blockIdx.x * 16;   // grid = H/16 = 256 tiles
  const int k0   = (threadIdx.x >> 5) * KC;
  v8f acc = wmma_gemv_partial<KC>(W_in, E_DIM, row0, k0, sx);
  reduce_store<true>(xi, row0, acc, b_in, sred);
}

// ---- Stage 2: partial GEMVs for the gates (dominant stage, 512 MB) ----
// blocks [0, 1024)      : gih[tile] = W_ih[tile rows] @ xi
// blocks [1024, 2048)   : ghh[tile] = W_hh[tile rows] @ h0
__global__ void __launch_bounds__(32 * NWAVES)
k_gates_part(const float* __restrict__ xi, const float* __restrict__ h0,
             const float* __restrict__ W_ih, const float* __restrict__ W_hh,
             float* __restrict__ gih, float* __restrict__ ghh) {
  __shared__ float sv[H_DIM];
  __shared__ float sred[NWAVES * 16];
  const bool is_hh = blockIdx.x >= (4 * H_DIM / 16);  // uniform branch
  const float* vecg = is_hh ? h0 : xi;
  const float* W    = is_hh ? W_hh : W_ih;
  float* dst        = is_hh ? ghh : gih;
  for (int i = threadIdx.x; i < H_DIM; i += 32 * NWAVES) sv[i] = vecg[i];
  __syncthreads();
  constexpr int KC = H_DIM / NWAVES;  // 512
  const int tile = blockIdx.x & ((4 * H_DIM / 16) - 1);
  const int row0 = tile * 16;
  const int k0   = (threadIdx.x >> 5) * KC;
  v8f acc = wmma_gemv_partial<KC>(W, H_DIM, row0, k0, sv);
  reduce_store<false>(dst, row0, acc, nullptr, sred);
}

// ---- Stage 3: combine partials + biases, LSTM cell (gate order i,f,g,o) ----
__global__ void __launch_bounds__(256)
k_cell(const float* __restrict__ gih, const float* __restrict__ ghh,
       const float* __restrict__ b_ih, const float* __restrict__ b_hh,
       const float* __restrict__ c0, float* __restrict__ h1,
       float* __restrict__ c1) {
  const int i = blockIdx.x * 256 + threadIdx.x;
  if (i < H_DIM) {
    const float ig = gih[i] + ghh[i] + b_ih[i] + b_hh[i];
    const float fg = gih[i + H_DIM] + ghh[i + H_DIM] +
                     b_ih[i + H_DIM] + b_hh[i + H_DIM];
    const float gg = gih[i + 2 * H_DIM] + ghh[i + 2 * H_DIM] +
                     b_ih[i + 2 * H_DIM] + b_hh[i + 2 * H_DIM];
    const float og = gih[i + 3 * H_DIM] + ghh[i + 3 * H_DIM] +
                     b_ih[i + 3 * H_DIM] + b_hh[i + 3 * H_DIM];
    const float si = 1.0f / (1.0f + expf(-ig));
    const float sf = 1.0f / (1.0f + expf(-fg));
    const float so = 1.0f / (1.0f + expf(-og));
    const float c  = sf * c0[i] + si * tanhf(gg);
    c1[i] = c;
    h1[i] = so * tanhf(c);
  }
}

// ---- Stage 4: out = W_out @ h1 + b_out (E rows, K=H) ----
__global__ void __launch_bounds__(32 * NWAVES)
k_out_proj(const float* __restrict__ h1, const float* __restrict__ W_out,
           const float* __restrict__ b_out, float* __restrict__ out) {
  __shared__ float sh[H_DIM];
  __shared__ float sred[NWAVES * 16];
  for (int i = threadIdx.x; i < H_DIM; i += 32 * NWAVES) sh[i] = h1[i];
  __syncthreads();
  constexpr int KC = H_DIM / NWAVES;  // 512
  const int row0 = blockIdx.x * 16;   // grid = E/16 = 64 tiles
  const int k0   = (threadIdx.x >> 5) * KC;
  v8f acc = wmma_gemv_partial<KC>(W_out, H_DIM, row0, k0, sh);
  reduce_store<true>(out, row0, acc, b_out, sred);
}

extern "C" void kernel_launch(void* const* d_in, const int* in_sizes, int n_in,
                              void* d_out, int out_size, void* d_ws,
                              size_t ws_size, hipStream_t stream) {
  (void)in_sizes; (void)n_in; (void)out_size; (void)ws_size;
  const float* x     = (const float*)d_in[0];   // (1, E)
  const float* h0    = (const float*)d_in[1];   // (1, 1, H)
  const float* c0    = (const float*)d_in[2];   // (1, 1, H)
  const float* W_in  = (const float*)d_in[3];   // (H, E)
  const float* b_in  = (const float*)d_in[4];   // (H,)
  const float* W_ih  = (const float*)d_in[5];   // (4H, H)
  const float* W_hh  = (const float*)d_in[6];   // (4H, H)
  const float* b_ih  = (const float*)d_in[7];   // (4H,)
  const float* b_hh  = (const float*)d_in[8];   // (4H,)
  const float* W_out = (const float*)d_in[9];   // (E, H)
  const float* b_out = (const float*)d_in[10];  // (E,)

  float* out = (float*)d_out;      // [0, 1024): out
  float* h1  = out + E_DIM;        // [1024, 5120): h1
  float* c1  = h1 + H_DIM;         // [5120, 9216): c1

  float* xi  = (float*)d_ws;       // H floats
  float* gih = xi + H_DIM;         // 4H floats: W_ih @ xi partial
  float* ghh = gih + 4 * H_DIM;    // 4H floats: W_hh @ h0 partial

  k_in_proj   <<<H_DIM / 16, 32 * NWAVES, 0, stream>>>(x, W_in, b_in, xi);
  k_gates_part<<<2 * (4 * H_DIM / 16), 32 * NWAVES, 0, stream>>>(
      xi, h0, W_ih, W_hh, gih, ghh);
  k_cell      <<<H_DIM / 256, 256, 0, stream>>>(gih, ghh, b_ih, b_hh, c0, h1, c1);
  k_out_proj  <<<E_DIM / 16, 32 * NWAVES, 0, stream>>>(h1, W_out, b_out, out);
}